// input_backbone_18760417149532
// MI455X (gfx1250) — compile-verified
//
#include <hip/hip_runtime.h>
#include <stdint.h>

// ---------------------------------------------------------------------------
// Submanifold sparse conv (27x [N,4]x[4,16]) + BatchNorm(train) + ReLU
// Core matrix op: V_WMMA_F32_16X16X4_F32  (K=4 == CIN, full fp32 precision)
// Two-phase tile body: (1) 27 independent hash-probe gathers in flight,
// (2) 27 back-to-back WMMA accumulates fed from LDS-staged weights.
// ---------------------------------------------------------------------------

typedef __attribute__((ext_vector_type(2))) float v2f;
typedef __attribute__((ext_vector_type(8))) float v8f;

#define GZ 41
#define GY 1600
#define GX 1408
#define BN_EPS 1e-3f

#define TBITS 20u
#define TSIZE (1u << TBITS)
#define TMASK (TSIZE - 1u)
#define EMPTY64 0xFFFFFFFFFFFFFFFFull

__device__ __forceinline__ uint32_t hslot(uint32_t key) {
    return (key * 2654435761u) >> (32u - TBITS);   // Fibonacci hash, top bits
}

// ---- kernel 1: clear hash table (ws is poisoned / stale between calls) ----
__global__ void ht_clear_kernel(unsigned long long* __restrict__ tab) {
    uint32_t i = (blockIdx.x * blockDim.x + threadIdx.x) * 8u;
    #pragma unroll
    for (int j = 0; j < 8; ++j) tab[i + j] = EMPTY64;
}

// ---- kernel 2: insert (key -> row).  Keys are unique, so any slot race
// resolution yields a table with identical lookup semantics. ----------------
__global__ void ht_build_kernel(const int* __restrict__ coords,
                                unsigned long long* __restrict__ tab, int N) {
    int n = blockIdx.x * blockDim.x + threadIdx.x;
    if (n >= N) return;
    const int4 c4 = *(const int4*)(coords + n * 4);
    uint32_t key = ((uint32_t)c4.y * GY + (uint32_t)c4.z) * GX + (uint32_t)c4.w;
    unsigned long long pack = ((unsigned long long)key << 32) | (uint32_t)n;
    uint32_t slot = hslot(key);
    #pragma unroll 1
    for (uint32_t p = 0; p < TSIZE; ++p) {
        unsigned long long old = atomicCAS(&tab[slot], EMPTY64, pack);
        if (old == EMPTY64) return;
        slot = (slot + 1u) & TMASK;
    }
}

__device__ __forceinline__ int ht_lookup(const unsigned long long* __restrict__ tab,
                                         uint32_t key) {
    uint32_t slot = hslot(key);
    #pragma unroll 1
    for (uint32_t p = 0; p < TSIZE; ++p) {
        unsigned long long e = tab[slot];
        if (e == EMPTY64) return -1;
        if ((uint32_t)(e >> 32) == key) return (int)(uint32_t)e;
        slot = (slot + 1u) & TMASK;
    }
    return -1;
}

// ---- kernel 3: gather + 27x WMMA f32 16x16x4 + per-block BN partials ------
// Wave tile: 16 voxels (M) x 16 couts (N).  Lane L:  m = L&15, khalf = L>>4.
//   A 16x4 layout:  v0 = A[m][2*khalf],   v1 = A[m][2*khalf+1]
//   B 4x16 layout:  v0 = W[2*khalf][col], v1 = W[2*khalf+1][col], col = m
//   C 16x16 layout: acc[i] = C[i + 8*khalf][m]
__launch_bounds__(256)
__global__ void subm_conv_kernel(const float* __restrict__ feats,
                                 const int*   __restrict__ coords,
                                 const float* __restrict__ weight,
                                 const unsigned long long* __restrict__ tab,
                                 float* __restrict__ out,
                                 float* __restrict__ partial,
                                 int N) {
    __shared__ float w_lds[27 * 64];
    __shared__ float red_s[256];
    __shared__ float red_q[256];

    const int tid = threadIdx.x;
    for (int i = tid; i < 27 * 64; i += 256) w_lds[i] = weight[i];
    __syncthreads();

    const int wv    = tid >> 5;
    const int lane  = tid & 31;
    const int m     = lane & 15;
    const int khalf = lane >> 4;          // 0 -> K{0,1}, 1 -> K{2,3}
    const int tile  = blockIdx.x * 8 + wv;
    const int row   = tile * 16 + m;
    const bool valid = row < N;

    int z = 0, y = 0, x = 0;
    if (valid) {
        const int4 c4 = *(const int4*)(coords + row * 4);
        z = c4.y; y = c4.z; x = c4.w;
    }

    // ---- phase 1: all 27 gathers in flight (compile-time stencil offsets) -
    v2f av[27];
    #pragma unroll
    for (int k = 0; k < 27; ++k) {
        const int dz = k / 9 - 1;              // folds to constants after
        const int dy = (k / 3) % 3 - 1;        // full unroll
        const int dx = k % 3 - 1;
        v2f a; a.x = 0.f; a.y = 0.f;
        if (k == 13) {
            // center tap: neighbor is the voxel itself — no hash probe
            if (valid) a = *(const v2f*)(feats + row * 4 + khalf * 2);
        } else if (valid) {
            const int nz = z + dz, ny = y + dy, nx = x + dx;
            if ((uint32_t)nz < GZ && (uint32_t)ny < GY && (uint32_t)nx < GX) {
                const uint32_t key =
                    ((uint32_t)nz * GY + (uint32_t)ny) * GX + (uint32_t)nx;
                const int ridx = ht_lookup(tab, key);
                if (ridx >= 0)
                    a = *(const v2f*)(feats + ridx * 4 + khalf * 2); // 8B-aligned
            }
        }
        av[k] = a;
    }

    // ---- phase 2: 27 chained WMMA accumulates, B frags streamed from LDS --
    v8f acc = {};
    #pragma unroll
    for (int k = 0; k < 27; ++k) {
        v2f b;
        b.x = w_lds[k * 64 + (khalf * 2 + 0) * 16 + m];
        b.y = w_lds[k * 64 + (khalf * 2 + 1) * 16 + m];
        acc = __builtin_amdgcn_wmma_f32_16x16x4_f32(
            /*neg_a=*/false, av[k], /*neg_b=*/false, b,
            /*c_mod=*/(short)0, acc, /*reuse_a=*/false, /*reuse_b=*/false);
    }

    // store conv result + local BN moments (channel of this lane is `m`)
    float s = 0.f, q = 0.f;
    #pragma unroll
    for (int i = 0; i < 8; ++i) {
        const int r = tile * 16 + i + 8 * khalf;
        const float v = acc[i];
        s += v; q += v * v;                 // zero rows (r>=N) contribute 0
        if (r < N) out[r * 16 + m] = v;
    }
    red_s[tid] = s;
    red_q[tid] = q;
    __syncthreads();

    // deterministic fixed-order block reduction (no float atomics)
    if (tid < 16) {
        float ss = 0.f, qq = 0.f;
        #pragma unroll
        for (int w = 0; w < 8; ++w) {
            ss += red_s[w * 32 + tid] + red_s[w * 32 + 16 + tid];
            qq += red_q[w * 32 + tid] + red_q[w * 32 + 16 + tid];
        }
        partial[blockIdx.x * 32 + tid]      = ss;
        partial[blockIdx.x * 32 + 16 + tid] = qq;
    }
}

// ---- kernel 4: fold partials -> per-channel affine (a, b) -----------------
__global__ void bn_stats_kernel(const float* __restrict__ partial,
                                const float* __restrict__ gamma,
                                const float* __restrict__ beta,
                                float* __restrict__ stats, int nblk, int N) {
    const int t = threadIdx.x;
    if (t >= 16) return;
    float s = 0.f, q = 0.f;
    #pragma unroll 1
    for (int b = 0; b < nblk; ++b) {       // fixed order -> deterministic
        s += partial[b * 32 + t];
        q += partial[b * 32 + 16 + t];
    }
    const float invN = 1.0f / (float)N;
    const float mu  = s * invN;
    const float var = q * invN - mu * mu;
    const float a   = rsqrtf(var + BN_EPS) * gamma[t];
    stats[t]      = a;
    stats[16 + t] = beta[t] - mu * a;
}

// ---- kernel 5: y = relu(x*a + b) ------------------------------------------
__global__ void bn_relu_kernel(float* __restrict__ out,
                               const float* __restrict__ stats, int total) {
    const int i = blockIdx.x * blockDim.x + threadIdx.x;
    if (i >= total) return;
    const int ch = i & 15;
    const float v = out[i] * stats[ch] + stats[16 + ch];
    out[i] = v > 0.f ? v : 0.f;
}

// ---------------------------------------------------------------------------
extern "C" void kernel_launch(void* const* d_in, const int* in_sizes, int n_in,
                              void* d_out, int out_size, void* d_ws, size_t ws_size,
                              hipStream_t stream) {
    const float* feats  = (const float*)d_in[0];
    const int*   coords = (const int*)  d_in[1];
    const float* weight = (const float*)d_in[2];
    const float* gamma  = (const float*)d_in[3];
    const float* beta   = (const float*)d_in[4];
    float* out = (float*)d_out;

    const int N = in_sizes[0] / 4;                     // [N, CIN=4]
    const int tiles = (N + 15) / 16;
    const int nblk  = (tiles + 7) / 8;                 // 8 waves (tiles) / block

    // workspace layout: [hash table 8MB][partials nblk*32 f32][stats 32 f32]
    unsigned long long* tab = (unsigned long long*)d_ws;
    float* partial = (float*)((char*)d_ws + (size_t)TSIZE * 8u);
    float* stats   = (float*)((char*)d_ws + (size_t)TSIZE * 8u
                                          + (size_t)nblk * 32u * sizeof(float));

    ht_clear_kernel<<<TSIZE / (256 * 8), 256, 0, stream>>>(tab);
    ht_build_kernel<<<(N + 255) / 256, 256, 0, stream>>>(coords, tab, N);
    subm_conv_kernel<<<nblk, 256, 0, stream>>>(feats, coords, weight, tab,
                                               out, partial, N);
    bn_stats_kernel<<<1, 64, 0, stream>>>(partial, gamma, beta, stats, nblk, N);
    const int total = N * 16;
    bn_relu_kernel<<<(total + 255) / 256, 256, 0, stream>>>(out, stats, total);
}